// AttentionCritic_26568667693820
// MI455X (gfx1250) — compile-verified
//
#include <hip/hip_runtime.h>
#include <math.h>

typedef float v2f __attribute__((ext_vector_type(2)));
typedef float v8f __attribute__((ext_vector_type(8)));
typedef unsigned int u32x4 __attribute__((ext_vector_type(4)));
typedef int i32x4 __attribute__((ext_vector_type(4)));
typedef int i32x8 __attribute__((ext_vector_type(8)));

#define BM 128
#define BN 64
#define BK 32
#define LDSTRIDE 36  // 32 data dwords + 4 pad dwords (TDM pad_interval=32dw, pad_amount=4dw)

// ---------------------------------------------------------------------------
// TDM: DMA a 2-D tile (tile_d1 rows x 32 floats) from a row-major tensor
// [tensor_d1 x tensor_d0] into LDS with 4-dword padding after each 32 dwords
// (-> LDS row stride of 36 floats). One instruction moves the whole panel;
// completion tracked with TENSORcnt.
// ---------------------------------------------------------------------------
__device__ __forceinline__
void tdm_load_panel(unsigned lds_off, const float* gptr,
                    int tensor_d0, int tensor_d1, int tile_d1) {
  const unsigned long long ga = (unsigned long long)(uintptr_t)gptr;
  const unsigned d0 = (unsigned)tensor_d0;
  const unsigned d1 = (unsigned)tensor_d1;

  u32x4 g0;
  g0[0] = 1u;                                   // count=1 (valid), user mode
  g0[1] = lds_off;                              // lds_addr (bytes)
  g0[2] = (unsigned)ga;                         // global_addr[31:0]
  g0[3] = (unsigned)((ga >> 32) & 0x01FFFFFFu)  // global_addr[56:32]
        | (2u << 30);                           // type=2 ("image")

  i32x8 g1;
  g1[0] = (int)((2u << 16)      // data_size = 4 bytes
              | (1u << 20)      // pad_enable
              | (4u << 22)      // pad_interval: 32 dwords
              | (3u << 25));    // pad_amount : 4 dwords
  g1[1] = (int)((d0 & 0xFFFFu) << 16);                          // tensor_dim0[15:0]
  g1[2] = (int)(((d0 >> 16) & 0xFFFFu) | ((d1 & 0xFFFFu) << 16)); // dim0 hi | dim1 lo
  g1[3] = (int)(((d1 >> 16) & 0xFFFFu) | ((unsigned)BK << 16)); // dim1 hi | tile_dim0=32
  g1[4] = (int)((unsigned)tile_d1 & 0xFFFFu);                   // tile_dim1 (tile_dim2=0)
  g1[5] = (int)d0;                                              // tensor_dim0_stride lo
  g1[6] = 0;                                                    // stride0 hi | stride1 lo
  g1[7] = 0;                                                    // stride1 hi

  const i32x4 gz4 = {0, 0, 0, 0};
  const i32x8 gz8 = {0, 0, 0, 0, 0, 0, 0, 0};
  // 6-arg toolchain form: (group0, group1, group2, group3, group4, cpol)
  __builtin_amdgcn_tensor_load_to_lds(g0, g1, gz4, gz4, gz8, 0);
}

// ---------------------------------------------------------------------------
// Fused WMMA fp32 GEMM:  OUT[M,N] = act( X[M,K] @ W[N,K]^T + bias[N] )
// ACT: 1 = tanh, 2 = sigmoid + write attention and state_attention
// Block tile 128x64, 8 waves (wave32), wave owns a 32x32 patch = 2x2 tiles of
// v_wmma_f32_16x16x4_f32. Panels staged by the Tensor Data Mover, double
// buffered so the DMA of panel k+1 overlaps WMMA on panel k.
// ---------------------------------------------------------------------------
template<int ACT>
__global__ __launch_bounds__(256)
void gemm_wmma_f32(const float* __restrict__ X, const float* __restrict__ W,
                   const float* __restrict__ bias, float* __restrict__ OUT,
                   const float* __restrict__ obs, float* __restrict__ sa,
                   int M, int N, int K) {
  __shared__ float As[2][BM][LDSTRIDE];  // activations, padded rows
  __shared__ float Bs[2][BN][LDSTRIDE];  // weight rows (untransposed), padded

  const int tid  = threadIdx.x;
  const int lane = tid & 31;
  const int wave = tid >> 5;   // 0..7
  const int wm   = wave & 3;   // row strip (4 strips of 32 rows)
  const int wn   = wave >> 2;  // col strip (2 strips of 32 cols)
  const int lm   = lane & 15;
  const int lh   = lane >> 4;  // half-wave select (ISA 7.12.2 layouts)

  const int row0 = blockIdx.y * BM;
  const int col0 = blockIdx.x * BN;

  const v8f vzero = {0.f, 0.f, 0.f, 0.f, 0.f, 0.f, 0.f, 0.f};
  v8f acc[2][2];
  acc[0][0] = vzero; acc[0][1] = vzero; acc[1][0] = vzero; acc[1][1] = vzero;

  // Prologue: DMA panel 0 into buffer 0.
  if (wave == 0) {
    tdm_load_panel((unsigned)(uintptr_t)&As[0][0][0],
                   X + (size_t)row0 * K, K, M, BM);
    tdm_load_panel((unsigned)(uintptr_t)&Bs[0][0][0],
                   W + (size_t)col0 * K, K, N, BN);
  }

  int buf = 0;
  for (int k0 = 0; k0 < K; k0 += BK) {
    const bool has_next = (k0 + BK) < K;
    if (wave == 0) {
      if (has_next) {
        // DMA next panel into the other buffer; overlaps with compute below.
        tdm_load_panel((unsigned)(uintptr_t)&As[buf ^ 1][0][0],
                       X + (size_t)row0 * K + (k0 + BK), K, M, BM);
        tdm_load_panel((unsigned)(uintptr_t)&Bs[buf ^ 1][0][0],
                       W + (size_t)col0 * K + (k0 + BK), K, N, BN);
        __builtin_amdgcn_s_wait_tensorcnt(2);  // current panel's 2 DMAs done
      } else {
        __builtin_amdgcn_s_wait_tensorcnt(0);
      }
    }
    __syncthreads();  // panel `buf` visible to all waves

#pragma unroll
    for (int kk = 0; kk < BK; kk += 4) {
      const int ka = kk + 2 * lh;
      v2f a[2], b[2];
      // A 16x4 f32 fragment: lane m<16 holds K={0,1}, lane m+16 holds K={2,3}
      a[0].x = As[buf][wm * 32 + lm][ka];       a[0].y = As[buf][wm * 32 + lm][ka + 1];
      a[1].x = As[buf][wm * 32 + 16 + lm][ka];  a[1].y = As[buf][wm * 32 + 16 + lm][ka + 1];
      // B 4x16 f32 fragment from untransposed W rows (k-contiguous per lane)
      b[0].x = Bs[buf][wn * 32 + lm][ka];       b[0].y = Bs[buf][wn * 32 + lm][ka + 1];
      b[1].x = Bs[buf][wn * 32 + 16 + lm][ka];  b[1].y = Bs[buf][wn * 32 + 16 + lm][ka + 1];
#pragma unroll
      for (int i = 0; i < 2; ++i)
#pragma unroll
        for (int j = 0; j < 2; ++j)
          acc[i][j] = __builtin_amdgcn_wmma_f32_16x16x4_f32(
              false, a[i], false, b[j], (short)0, acc[i][j], false, false);
    }
    __syncthreads();  // all waves done with `buf` before it is re-filled
    buf ^= 1;
  }

  // --- epilogue: bias + activation (+ fused state-attention for ACT==2) ---
#pragma unroll
  for (int i = 0; i < 2; ++i) {
#pragma unroll
    for (int j = 0; j < 2; ++j) {
      const int gcol = col0 + wn * 32 + j * 16 + lm;
      const float bv = bias[gcol];
#pragma unroll
      for (int r = 0; r < 8; ++r) {
        const int grow = row0 + wm * 32 + i * 16 + 8 * lh + r;  // C/D: M = r + 8*lh
        const float val = acc[i][j][r] + bv;
        const size_t idx = (size_t)grow * N + gcol;
        if constexpr (ACT == 1) {
          OUT[idx] = tanhf(val);
        } else if constexpr (ACT == 2) {
          const float att = 1.0f / (1.0f + __expf(-val));  // softmax(2)[1] == sigmoid(l1-l0)
          OUT[idx] = att;                                  // attention -> d_out
          sa[idx] = obs[idx] * att;                        // state_attention
        } else {
          OUT[idx] = val;
        }
      }
    }
  }
}

// ---------------------------------------------------------------------------
// Fold each 2-logit attention head into a single difference column:
// softmax([l0,l1])[1] == sigmoid(l1 - l0)  =>  Wd = W[:, :, 1] - W[:, :, 0]
// ---------------------------------------------------------------------------
__global__ __launch_bounds__(256)
void prep_attn(const float* __restrict__ attn_W, const float* __restrict__ attn_b,
               float* __restrict__ Wd, float* __restrict__ bd) {
  const int i = blockIdx.x * blockDim.x + threadIdx.x;  // 0 .. 1024*512-1
  const float2 w = ((const float2*)attn_W)[i];
  Wd[i] = w.y - w.x;
  if (i < 1024) {
    const float2 b = ((const float2*)attn_b)[i];
    bd[i] = b.y - b.x;
  }
}

// ---------------------------------------------------------------------------
// v = h2 @ W3^T + b3 : wave-per-row float4 dot + shuffle reduction
// ---------------------------------------------------------------------------
__global__ __launch_bounds__(256)
void gemv_out(const float* __restrict__ H, const float* __restrict__ W3,
              const float* __restrict__ b3, float* __restrict__ out, int K) {
  const int wave = threadIdx.x >> 5;
  const int lane = threadIdx.x & 31;
  const int row = blockIdx.x * 8 + wave;
  const float4* h = (const float4*)(H + (size_t)row * K);
  const float4* w = (const float4*)W3;
  float s = 0.f;
  const int n4 = K >> 2;
  for (int i = lane; i < n4; i += 32) {
    const float4 a = h[i];
    const float4 b = w[i];
    s += a.x * b.x + a.y * b.y + a.z * b.z + a.w * b.w;
  }
#pragma unroll
  for (int off = 16; off > 0; off >>= 1) s += __shfl_xor(s, off, 32);
  if (lane == 0) out[row] = s + b3[0];
}

// ---------------------------------------------------------------------------
extern "C" void kernel_launch(void* const* d_in, const int* in_sizes, int n_in,
                              void* d_out, int out_size, void* d_ws, size_t ws_size,
                              hipStream_t stream) {
  (void)in_sizes; (void)n_in; (void)out_size; (void)ws_size;

  const float* obs    = (const float*)d_in[0];   // [4096,1024]
  const float* We     = (const float*)d_in[1];   // [512,1024]
  const float* be     = (const float*)d_in[2];   // [512]
  const float* attn_W = (const float*)d_in[3];   // [1024,512,2]
  const float* attn_b = (const float*)d_in[4];   // [1024,2]
  const float* W1     = (const float*)d_in[5];   // [2048,1024]
  const float* b1     = (const float*)d_in[6];
  const float* W2     = (const float*)d_in[7];   // [2048,2048]
  const float* b2     = (const float*)d_in[8];
  const float* W3     = (const float*)d_in[9];   // [1,2048]
  const float* b3     = (const float*)d_in[10];

  float* out_v   = (float*)d_out;        // [4096]
  float* out_att = out_v + 4096;         // [4096,1024]

  float* ws = (float*)d_ws;
  float* e  = ws;                        // 4096*512
  float* Wd = e  + 4096 * 512;           // 1024*512
  float* bd = Wd + 1024 * 512;           // 1024
  float* sa = bd + 1024;                 // 4096*1024
  float* h1 = sa + 4096 * 1024;          // 4096*2048
  float* h2 = h1 + 4096 * 2048;          // 4096*2048

  // Fold attention heads (2 logits -> 1 difference column)
  prep_attn<<<(1024 * 512) / 256, 256, 0, stream>>>(attn_W, attn_b, Wd, bd);

  // Stage 1: e = tanh(obs @ We^T + be)         M=4096 N=512  K=1024
  gemm_wmma_f32<1><<<dim3(512 / BN, 4096 / BM), 256, 0, stream>>>(
      obs, We, be, e, nullptr, nullptr, 4096, 512, 1024);

  // Stage 2: attention = sigmoid(e @ Wd^T + bd); sa = obs * attention
  gemm_wmma_f32<2><<<dim3(1024 / BN, 4096 / BM), 256, 0, stream>>>(
      e, Wd, bd, out_att, obs, sa, 4096, 1024, 512);

  // Stage 3: h1 = tanh(sa @ W1^T + b1)         M=4096 N=2048 K=1024
  gemm_wmma_f32<1><<<dim3(2048 / BN, 4096 / BM), 256, 0, stream>>>(
      sa, W1, b1, h1, nullptr, nullptr, 4096, 2048, 1024);

  // Stage 4: h2 = tanh(h1 @ W2^T + b2)         M=4096 N=2048 K=2048
  gemm_wmma_f32<1><<<dim3(2048 / BN, 4096 / BM), 256, 0, stream>>>(
      h1, W2, b2, h2, nullptr, nullptr, 4096, 2048, 2048);

  // Stage 5: v = h2 @ W3^T + b3
  gemv_out<<<4096 / 8, 256, 0, stream>>>(h2, W3, b3, out_v, 2048);
}